// CausalCentroidPyramid_46377056862839
// MI455X (gfx1250) — compile-verified
//
#include <hip/hip_runtime.h>
#include <hip/hip_bf16.h>

// Problem constants (match reference setup_inputs)
constexpr int Bn = 4;
constexpr int Tn = 4096;
constexpr int Cn = 512;
constexpr int Kn = 8;          // NUM_SCALES
constexpr float TAUf = 1e-6f;
constexpr float EPSf = 1e-12f;

// Segmented scan config
constexpr int NSEG = 32;
constexpr int SEGT = Tn / NSEG;       // 128 timesteps per segment
constexpr int BLKS = SEGT / 16;       // 8 WMMA blocks per segment
static_assert(NSEG * SEGT == Tn, "segment split");

typedef float v2f __attribute__((ext_vector_type(2)));
typedef float v4f __attribute__((ext_vector_type(4)));
typedef float v8f __attribute__((ext_vector_type(8)));

// ---------------------------------------------------------------------------
// Phase 1: per-segment local prefix sums via WMMA (S_blk = L16 * X_blk + carry).
// One wave per (b, 16-channel slab, segment). Register double-buffering keeps
// the next block's loads in flight while the current block's WMMAs execute.
// Writes the segment total (last row's prefix) for phase 2.
// ---------------------------------------------------------------------------
__global__ __launch_bounds__(32) void scan_seg_kernel(const float* __restrict__ x,
                                                      float* __restrict__ S,
                                                      float* __restrict__ segtot) {
  const int wid  = blockIdx.x;                 // Bn * 32 * NSEG waves
  const int seg  = wid % NSEG;
  const int cb   = (wid / NSEG) & 31;          // Cn/16 == 32 slabs
  const int b    = wid / (NSEG * 32);
  const int c0   = cb * 16;
  const int lane = threadIdx.x;                // wave32
  const int n    = lane & 15;                  // channel within slab; also M for A
  const int half = lane >> 4;

  const size_t base = (size_t)b * Tn * Cn;
  const float* xp = x + base + c0 + n;
  float* Sp = S + base + c0 + n;
  const int tbase = seg * SEGT;

  // Constant A operands: L16 (lower-triangular ones), columns [4g,4g+4).
  v2f A[4];
#pragma unroll
  for (int g = 0; g < 4; ++g) {
    const int k0 = 4 * g + 2 * half;
    A[g].x = (k0     <= n) ? 1.0f : 0.0f;
    A[g].y = (k0 + 1 <= n) ? 1.0f : 0.0f;
  }

  v2f buf[2][4];
  auto loadB = [&](int t0, v2f* Bv) {
#pragma unroll
    for (int g = 0; g < 4; ++g) {
      const int r0 = t0 + 4 * g + half;        // B: VGPR0 rows K=0/1, VGPR1 rows K=2/3
      Bv[g].x = __builtin_nontemporal_load(xp + (size_t)r0 * Cn);
      Bv[g].y = __builtin_nontemporal_load(xp + (size_t)(r0 + 2) * Cn);
    }
  };

  loadB(tbase, buf[0]);
  float carry = 0.0f;
#pragma unroll
  for (int blk = 0; blk < BLKS; ++blk) {
    if (blk + 1 < BLKS) loadB(tbase + 16 * (blk + 1), buf[(blk + 1) & 1]);
    v8f acc = {};
#pragma unroll
    for (int g = 0; g < 4; ++g)
      acc = __builtin_amdgcn_wmma_f32_16x16x4_f32(false, A[g], false, buf[blk & 1][g],
                                                  (short)0, acc, false, false);
#pragma unroll
    for (int r = 0; r < 8; ++r) acc[r] += carry;
    const int t0 = tbase + 16 * blk;
#pragma unroll
    for (int r = 0; r < 8; ++r)               // D layout: M = 8*half + r, N = n
      Sp[(size_t)(t0 + 8 * half + r) * Cn] = acc[r];
    carry = __shfl(acc[7], 16 + n, 32);       // row M=15 -> all lanes' channel n
  }
  if (half == 0)
    segtot[((size_t)b * NSEG + seg) * Cn + c0 + n] = carry;
}

// ---------------------------------------------------------------------------
// Phase 2: exclusive scan of segment totals per (b, c). 2048 threads, trivial.
// ---------------------------------------------------------------------------
__global__ __launch_bounds__(256) void seg_scan_kernel(const float* __restrict__ segtot,
                                                       float* __restrict__ segbase) {
  const int id = blockIdx.x * 256 + threadIdx.x;   // Bn*Cn
  const int b = id / Cn, c = id % Cn;
  float run = 0.0f;
  for (int s = 0; s < NSEG; ++s) {
    const size_t off = ((size_t)b * NSEG + s) * Cn + c;
    const float v = segtot[off];
    segbase[off] = run;
    run += v;
  }
}

// ---------------------------------------------------------------------------
// Phase 3: S[b,t,c] += segbase[b, t>>7, c].  L2-resident RMW (S = 32MB << 192MB).
// ---------------------------------------------------------------------------
__global__ __launch_bounds__(256) void fixup_kernel(float* __restrict__ S,
                                                    const float* __restrict__ segbase) {
  constexpr int C4 = Cn / 4;
  const size_t idx = (size_t)blockIdx.x * 256 + threadIdx.x;  // Bn*Tn*C4 lanes
  const int c4 = (int)(idx % C4);
  const int t  = (int)((idx / C4) % Tn);
  const int b  = (int)(idx / ((size_t)C4 * Tn));
  const int s  = t >> 7;                                      // SEGT == 128
  float4* S4 = reinterpret_cast<float4*>(S);
  const float4* B4 = reinterpret_cast<const float4*>(segbase);
  float4 v = S4[idx];
  const float4 a = B4[((size_t)b * NSEG + s) * C4 + c4];
  v.x += a.x; v.y += a.y; v.z += a.z; v.w += a.w;
  S4[idx] = v;
}

// ---------------------------------------------------------------------------
// Phase 4: phase transport for 8 dyadic scales from prefix sums.
// One wave per t (8 waves / block). Lane owns 16 channels.
//   curr_k = (P(t)      - P(t-2^k)    )/2^k  if t >= 2^k-1      else 0
//   prev_k = (P(t-2^k)  - P(t-2^{k+1}))/2^k  if t >= 2^{k+1}-1  else 0
//   out    = transport(curr,prev)            if t >= 2^k        else 0
// Output is streamed with NT stores so the 268MB write doesn't evict S from L2.
// ---------------------------------------------------------------------------
__device__ inline void loadrow16(const float* __restrict__ Pb, int pos, int ch0,
                                 float* v) {
  if (pos >= 0) {
    const float4* p = reinterpret_cast<const float4*>(Pb + (size_t)pos * Cn + ch0);
#pragma unroll
    for (int i = 0; i < 4; ++i) {
      float4 q = p[i];
      v[4 * i + 0] = q.x; v[4 * i + 1] = q.y; v[4 * i + 2] = q.z; v[4 * i + 3] = q.w;
    }
  } else {
#pragma unroll
    for (int i = 0; i < 16; ++i) v[i] = 0.0f;
  }
}

__global__ __launch_bounds__(256) void pyramid_kernel(const float* __restrict__ S,
                                                      float* __restrict__ out) {
  const int lane = threadIdx.x & 31;
  const int wav  = threadIdx.x >> 5;
  const int b    = blockIdx.x / (Tn / 8);
  const int t    = (blockIdx.x % (Tn / 8)) * 8 + wav;
  const int ch0  = lane * 16;

  const float* Pb = S + (size_t)b * Tn * Cn;
  float* Ob = out + ((size_t)b * Tn + t) * Kn * Cn;

  // Pull the next tile's P(t) row toward this WGP (global_prefetch_b8).
  if (t + 8 < Tn) __builtin_prefetch(Pb + (size_t)(t + 8) * Cn + ch0, 0, 1);

  float Pt[16], Arow[16], Brow[16];
  loadrow16(Pb, t,     ch0, Pt);     // P(t)
  loadrow16(Pb, t - 1, ch0, Arow);   // P(t - 2^0)

#pragma unroll
  for (int k = 0; k < Kn; ++k) {
    loadrow16(Pb, t - (2 << k), ch0, Brow);       // P(t - 2^{k+1})
    const float inv   = 1.0f / (float)(1 << k);
    const bool cur_ok = t >= (1 << k) - 1;
    const bool prv_ok = t >= (2 << k) - 1;

    float cu[16], pv[16];
    float cc = 0.0f, cp = 0.0f, pp = 0.0f;
#pragma unroll
    for (int j = 0; j < 16; ++j) {
      const float c_ = cur_ok ? (Pt[j]   - Arow[j]) * inv : 0.0f;
      const float p_ = prv_ok ? (Arow[j] - Brow[j]) * inv : 0.0f;
      cu[j] = c_; pv[j] = p_;
      cc += c_ * c_; cp += c_ * p_; pp += p_ * p_;
    }
#pragma unroll
    for (int off = 16; off > 0; off >>= 1) {      // wave32 butterflies
      cc += __shfl_xor(cc, off, 32);
      cp += __shfl_xor(cp, off, 32);
      pp += __shfl_xor(pp, off, 32);
    }

    const float ncr = sqrtf(cc), npr = sqrtf(pp);
    const float nc  = fmaxf(ncr, EPSf), np = fmaxf(npr, EPSf);
    const float c   = cp / (nc * np);
    const float a   = (cp - pp) / np;     // dot(v, w)
    const float bb  = (cc - cp) / nc;     // dot(u, w)
    const float den = fmaxf(1.0f + c, TAUf);
    const bool trivial = (c > 1.0f - TAUf) || (ncr < TAUf) || (npr < TAUf);
    const bool nneg    = (c < -1.0f + TAUf);

    // general branch: y = w + gc*curr + gp*prev
    const float gc = (-a + (a * c - bb) / den) / nc;
    const float gp = ( bb + (bb * c - a) / den) / np;

    // antipodal branch (wave-uniform, rare): rank-1 reflection terms
    float beta2 = 0.0f, gamma = 0.0f; int mch = -1;
    if (nneg) {
      float mv = fabsf(pv[0]); float mprev = pv[0], mcur = cu[0];
      mch = ch0;
#pragma unroll
      for (int j = 1; j < 16; ++j) {
        const float av = fabsf(pv[j]);
        if (av < mv) { mv = av; mch = ch0 + j; mprev = pv[j]; mcur = cu[j]; }
      }
#pragma unroll
      for (int off = 16; off > 0; off >>= 1) {
        const float ov  = __shfl_xor(mv, off, 32);
        const int   oc  = __shfl_xor(mch, off, 32);
        const float op  = __shfl_xor(mprev, off, 32);
        const float ocu = __shfl_xor(mcur, off, 32);
        if (ov < mv || (ov == mv && oc < mch)) { mv = ov; mch = oc; mprev = op; mcur = ocu; }
      }
      const float pvu   = mprev / np;            // e.v (v at argmin channel)
      const float widx  = mcur - mprev;          // w at argmin channel
      const float vv    = pp / (np * np);
      const float pn2   = 1.0f - 2.0f * pvu * pvu + pvu * pvu * vv;
      const float pnorm = fmaxf(sqrtf(fmaxf(pn2, 0.0f)), EPSf);
      const float pdw   = (widx - pvu * a) / pnorm;   // dot(p, w)
      beta2 = (-2.0f * a + 2.0f * pdw * pvu / pnorm) / np;
      gamma = 2.0f * pdw / pnorm;
    }

    const bool keep = (t >= (1 << k));
    float* orow = Ob + (size_t)k * Cn + ch0;
#pragma unroll
    for (int i = 0; i < 4; ++i) {
      v4f y4;
#pragma unroll
      for (int q = 0; q < 4; ++q) {
        const int j = 4 * i + q;
        const float w = cu[j] - pv[j];
        float y;
        if (nneg) {
          y = w + pv[j] * beta2 - ((ch0 + j) == mch ? gamma : 0.0f);
        } else if (trivial) {
          y = w;
        } else {
          y = w + cu[j] * gc + pv[j] * gp;
        }
        y4[q] = keep ? y : 0.0f;
      }
      __builtin_nontemporal_store(y4, reinterpret_cast<v4f*>(orow + 4 * i));
    }

#pragma unroll
    for (int j = 0; j < 16; ++j) Arow[j] = Brow[j];   // P(t-2^k) for next scale
  }
}

// ---------------------------------------------------------------------------
extern "C" void kernel_launch(void* const* d_in, const int* in_sizes, int n_in,
                              void* d_out, int out_size, void* d_ws, size_t ws_size,
                              hipStream_t stream) {
  const float* x = (const float*)d_in[0];        // (4,4096,512) f32
  float* out = (float*)d_out;                    // (4,4096,8,512) f32

  float* S       = (float*)d_ws;                             // 32 MB prefix sums
  float* segtot  = S + (size_t)Bn * Tn * Cn;                 // 256 KB
  float* segbase = segtot + (size_t)Bn * NSEG * Cn;          // 256 KB

  // Phase 1: WMMA local scans, 4096 waves (latency overlapped, double-buffered)
  scan_seg_kernel<<<dim3(Bn * 32 * NSEG), dim3(32), 0, stream>>>(x, S, segtot);
  // Phase 2: scan of segment totals
  seg_scan_kernel<<<dim3((Bn * Cn) / 256), dim3(256), 0, stream>>>(segtot, segbase);
  // Phase 3: add segment bases into S (L2-resident RMW)
  fixup_kernel<<<dim3((Bn * Tn * (Cn / 4)) / 256), dim3(256), 0, stream>>>(S, segbase);
  // Phase 4: 8-scale phase transport, streams 268MB once with NT stores
  pyramid_kernel<<<dim3(Bn * (Tn / 8)), dim3(256), 0, stream>>>(S, out);
}